// NeuralNeighborhoodFlow_25855703122108
// MI455X (gfx1250) — compile-verified
//
#include <hip/hip_runtime.h>
#include <math.h>

typedef float v2f __attribute__((ext_vector_type(2)));
typedef float v8f __attribute__((ext_vector_type(8)));

#define DIM 512
#define HID 2048
#define NBR 512
#define ROWS 513              // 1 (y) + neighbors
#define STATE (ROWS * DIM)    // 262656
#define T_STEPS 9
#define SUBSTEPS 2

// ---------------- GEMM config: 64x128 macro tile, 8 waves of 32x32 ----------
#define BM 64
#define BN 128
#define BK 32

// ---- CDNA5 async global->LDS copy (ASYNCcnt-tracked, no VGPR round trip) ---
__device__ __forceinline__ unsigned lds_off(const void* p) {
    // Shared aperture has zero low 32 bits -> low 32 bits of the flat address
    // of a __shared__ object are the wave-relative LDS byte offset (= VDST VGPR).
    return (unsigned)(uintptr_t)p;
}
__device__ __forceinline__ void async_ld_b128(unsigned ldst, const void* gaddr) {
    asm volatile("global_load_async_to_lds_b128 %0, %1, off"
                 :: "v"(ldst), "v"(gaddr) : "memory");
}
__device__ __forceinline__ void wait_async0() {
    asm volatile("s_wait_asynccnt 0x0" ::: "memory");
}

// C[M x N] = A[M x K] @ B[K x N], row-major fp32, via V_WMMA_F32_16X16X4_F32.
// Double-buffered LDS, async global->LDS staging overlapped with WMMA compute.
// mode 0: plain   mode 1: epilogue u=(1-a^2)*(s-a*s^2), a=aux[col]
// mode 2: row 0 gets +aux[col] (bias b2 for the f(y) row)
__global__ __launch_bounds__(256)
void gemm_wmma_f32(const float* __restrict__ A, int lda,
                   const float* __restrict__ B, int ldb,
                   float* __restrict__ C, int ldc,
                   int M, int K,
                   const float* __restrict__ aux, int mode)
{
    __shared__ float As[2][BM][BK + 4];   // pad -> stride 36 words
    __shared__ float Bs[2][BK][BN + 4];   // pad -> stride 132 words

    const int tid  = threadIdx.x;
    const int lane = tid & 31;
    const int wave = tid >> 5;
    const int wm   = wave >> 2;              // 0..1
    const int wn   = wave & 3;               // 0..3
    const int l16  = lane & 15;
    const int kb   = (lane >> 4) << 1;       // K sub-offset: 0 (lanes 0-15) / 2 (lanes 16-31)
    const int rowOff = (lane >> 4) << 3;     // C/D layout: +8 rows for lanes 16-31

    const int bm = blockIdx.y * BM;
    const int bn = blockIdx.x * BN;

    // per-thread staging coordinates (A: 2 x float4, B: 4 x float4 per tile)
    const int ar0 = tid >> 3, ar1 = (256 + tid) >> 3;          // A rows 0..63
    const int ac4 = (tid & 7) << 2;                            // A col 0..28
    // B: linear p*256+tid -> row = linear>>5 (0..31), col4 = (linear&31)<<2

    auto issue_tile = [&](int buf, int k0) {
        int gr0 = bm + ar0, gr1 = bm + ar1;
        if (gr0 < M) async_ld_b128(lds_off(&As[buf][ar0][ac4]),
                                   A + (size_t)gr0 * lda + k0 + ac4);
        else         *reinterpret_cast<float4*>(&As[buf][ar0][ac4]) = make_float4(0.f,0.f,0.f,0.f);
        if (gr1 < M) async_ld_b128(lds_off(&As[buf][ar1][ac4]),
                                   A + (size_t)gr1 * lda + k0 + ac4);
        else         *reinterpret_cast<float4*>(&As[buf][ar1][ac4]) = make_float4(0.f,0.f,0.f,0.f);
        #pragma unroll
        for (int p = 0; p < 4; ++p) {
            int linear = p * 256 + tid;
            int r  = linear >> 5;
            int c4 = (linear & 31) << 2;
            async_ld_b128(lds_off(&Bs[buf][r][c4]),
                          B + (size_t)(k0 + r) * ldb + bn + c4);
        }
    };

    v8f acc[2][2];
    #pragma unroll
    for (int m = 0; m < 2; ++m)
        #pragma unroll
        for (int n = 0; n < 2; ++n)
            acc[m][n] = (v8f){0.f, 0.f, 0.f, 0.f, 0.f, 0.f, 0.f, 0.f};

    issue_tile(0, 0);
    wait_async0();
    __syncthreads();

    const int nblk = K / BK;
    for (int blk = 0; blk < nblk; ++blk) {
        const int cur = blk & 1;
        if (blk + 1 < nblk)
            issue_tile(cur ^ 1, (blk + 1) * BK);   // overlap next tile with compute

        #pragma unroll
        for (int kk = 0; kk < BK; kk += 4) {
            v2f af[2], bf[2];
            #pragma unroll
            for (int m = 0; m < 2; ++m) {
                int r = wm * 32 + m * 16 + l16;      // A frag: 16x4, M striped on lanes
                af[m].x = As[cur][r][kk + kb];
                af[m].y = As[cur][r][kk + kb + 1];
            }
            #pragma unroll
            for (int n = 0; n < 2; ++n) {
                int c = wn * 32 + n * 16 + l16;      // B frag: 4x16, N striped on lanes
                bf[n].x = Bs[cur][kk + kb][c];
                bf[n].y = Bs[cur][kk + kb + 1][c];
            }
            #pragma unroll
            for (int m = 0; m < 2; ++m)
                #pragma unroll
                for (int n = 0; n < 2; ++n)
                    acc[m][n] = __builtin_amdgcn_wmma_f32_16x16x4_f32(
                        false, af[m], false, bf[n], (short)0, acc[m][n], false, false);
        }

        wait_async0();       // my async copies into the other buffer are done
        __syncthreads();     // everyone finished reading cur + finished their copies
    }

    // Epilogue per C/D layout: VGPR r holds (M = r + rowOff, N = l16)
    #pragma unroll
    for (int m = 0; m < 2; ++m) {
        #pragma unroll
        for (int n = 0; n < 2; ++n) {
            int gc = bn + wn * 32 + n * 16 + l16;
            #pragma unroll
            for (int r = 0; r < 8; ++r) {
                int gr = bm + wm * 32 + m * 16 + rowOff + r;
                if (gr < M) {
                    float s = acc[m][n][r];
                    if (mode == 1) {
                        float a = aux[gc];                       // a = tanh(h) at this hidden unit
                        s = (1.f - a * a) * (s - a * s * s);     // J·v + 0.5·H[v,v] pre-W2 factor
                    } else if (mode == 2) {
                        if (gr == 0) s += aux[gc];               // b2 on the f(y) row
                    }
                    C[(size_t)gr * ldc + gc] = s;
                }
            }
        }
    }
}

// h = ty @ W1 + b1 ; U[0][j] = tanh(h_j)   (ty = stage row 0, cached in LDS)
__global__ __launch_bounds__(256)
void hidden_kernel(const float* __restrict__ stage, const float* __restrict__ W1,
                   const float* __restrict__ b1, float* __restrict__ U)
{
    __shared__ float ty[DIM];
    int tid = threadIdx.x;
    for (int i = tid; i < DIM; i += 256) ty[i] = stage[i];
    __syncthreads();
    int j = blockIdx.x * 256 + tid;
    float h = b1[j];
    #pragma unroll 8
    for (int i = 0; i < DIM; ++i)
        h = fmaf(ty[i], W1[(size_t)i * HID + j], h);
    U[j] = tanhf(h);
}

// acc (=|+=) w*k ; stage = u + (cMul*dt)*k
__global__ __launch_bounds__(256)
void combine_kernel(const float* __restrict__ u, const float* __restrict__ k,
                    float* __restrict__ acc, float* __restrict__ stage,
                    const float* __restrict__ ts, int t, float cMul, float w, int init)
{
    int i = blockIdx.x * 256 + threadIdx.x;
    float dt = (ts[t + 1] - ts[t]) * (1.0f / SUBSTEPS);
    float kv = k[i];
    acc[i]   = init ? (w * kv) : fmaf(w, kv, acc[i]);
    stage[i] = fmaf(cMul * dt, kv, u[i]);
}

// u += dt/6 * (acc + k4) ; optionally emit trajectory sample
__global__ __launch_bounds__(256)
void final_kernel(float* __restrict__ u, const float* __restrict__ k,
                  const float* __restrict__ acc, const float* __restrict__ ts,
                  int t, float* __restrict__ out)
{
    int i = blockIdx.x * 256 + threadIdx.x;
    float dt = (ts[t + 1] - ts[t]) * (1.0f / SUBSTEPS);
    float nu = fmaf(dt * (1.0f / 6.0f), acc[i] + k[i], u[i]);
    u[i] = nu;
    if (out) out[i] = nu;
}

__global__ __launch_bounds__(256)
void init_kernel(const float* __restrict__ y0, const float* __restrict__ Dy0,
                 float* __restrict__ u, float* __restrict__ out)
{
    int i = blockIdx.x * 256 + threadIdx.x;
    float v = (i < DIM) ? y0[i] : Dy0[i - DIM];
    u[i] = v;
    out[i] = v;
}

extern "C" void kernel_launch(void* const* d_in, const int* in_sizes, int n_in,
                              void* d_out, int out_size, void* d_ws, size_t ws_size,
                              hipStream_t stream)
{
    (void)in_sizes; (void)n_in; (void)out_size; (void)ws_size;
    const float* ts  = (const float*)d_in[0];
    const float* y0  = (const float*)d_in[1];
    const float* Dy0 = (const float*)d_in[2];
    const float* W1  = (const float*)d_in[3];
    const float* b1  = (const float*)d_in[4];
    const float* W2  = (const float*)d_in[5];
    const float* b2  = (const float*)d_in[6];
    float* out = (float*)d_out;

    float* ws    = (float*)d_ws;
    float* u     = ws;                 // [513,512]  state: row0=y, rows1..512=Dy
    float* stage = u + STATE;          // [513,512]  RK4 stage input
    float* kbuf  = stage + STATE;      // [513,512]  rhs output: row0=f(y), rows=dDy
    float* accb  = kbuf + STATE;       // [513,512]  k1+2k2+2k3 accumulator
    float* U     = accb + STATE;       // [513,2048] row0=a=tanh(h), rows1..512=(1-a^2)(s-a s^2)

    const int EBLK = STATE / 256;      // 1026 exact

    init_kernel<<<EBLK, 256, 0, stream>>>(y0, Dy0, u, out);

    const float cMul[3] = {0.5f, 0.5f, 1.0f};
    const float wAcc[3] = {1.0f, 2.0f, 2.0f};

    dim3 g1(HID / BN, (NBR + BM - 1) / BM);      // 16 x 8  : S/U = Dy@W1
    dim3 g2(DIM / BN, (ROWS + BM - 1) / BM);     // 4 x 9   : [a;U]@W2 -> k

    for (int t = 0; t < T_STEPS - 1; ++t) {
        for (int sub = 0; sub < SUBSTEPS; ++sub) {
            for (int s = 0; s < 4; ++s) {
                const float* sp = (s == 0) ? u : stage;
                hidden_kernel<<<HID / 256, 256, 0, stream>>>(sp, W1, b1, U);
                gemm_wmma_f32<<<g1, 256, 0, stream>>>(sp + DIM, DIM, W1, HID,
                                                      U + HID, HID, NBR, DIM, U, 1);
                gemm_wmma_f32<<<g2, 256, 0, stream>>>(U, HID, W2, DIM,
                                                      kbuf, DIM, ROWS, HID, b2, 2);
                if (s < 3)
                    combine_kernel<<<EBLK, 256, 0, stream>>>(u, kbuf, accb, stage,
                                                             ts, t, cMul[s], wAcc[s], s == 0);
            }
            float* outPtr = (sub == SUBSTEPS - 1) ? (out + (size_t)(t + 1) * STATE) : nullptr;
            final_kernel<<<EBLK, 256, 0, stream>>>(u, kbuf, accb, ts, t, outPtr);
        }
    }
}